// GCNModel_10960756540204
// MI455X (gfx1250) — compile-verified
//
#include <hip/hip_runtime.h>

// ---------------------------------------------------------------------------
// 2-layer GCN (PyG GCNConv semantics) for MI455X / gfx1250.
//   h  = relu( scatter_add(norm * (x @ W1)[src] -> dst) + b1 )
//   out =      scatter_add(norm * (h @ W2)[src] -> dst) + b2
// GEMMs run on V_WMMA_F32_16X16X4_F32 (exact fp32 matrix cores), with the
// A-tile staged in LDS (shared by all 8 waves of a block) and the weights
// repacked into WMMA-fragment-major order for single-b64 coalesced loads.
// ---------------------------------------------------------------------------

typedef float v2f __attribute__((ext_vector_type(2)));
typedef float v8f __attribute__((ext_vector_type(8)));

#define N_NODES 50000
#define N_EDGES 800000
#define IN_CH   512
#define HID_CH  256
#define OUT_CH  128

// ---------------------------------------------------------------- utilities
__global__ void gcn_fill(float* __restrict__ p, float v, int n) {
    int i = blockIdx.x * blockDim.x + threadIdx.x;
    if (i < n) p[i] = v;
}

__global__ void gcn_deg_accum(const long long* __restrict__ dst,
                              float* __restrict__ deg) {
    int e = blockIdx.x * blockDim.x + threadIdx.x;
    if (e < N_EDGES) atomicAdd(&deg[(int)dst[e]], 1.0f);
}

__global__ void gcn_rsqrt(float* __restrict__ p, int n) {
    int i = blockIdx.x * blockDim.x + threadIdx.x;
    if (i < n) p[i] = rsqrtf(p[i]);   // deg >= 1 always (self-loops)
}

// ------------------------------------------------ weight repack (frag-major)
// Bp is organized as [ntile][kstep][lane] float2 records (256 B per kstep):
//   lane = hi*16 + n16, fragment = { B[ks*4 + hi*2][nt*16+n16],
//                                    B[ks*4 + hi*2 + 1][nt*16+n16] }
// so each wave's B fragment for one WMMA is a single contiguous 256 B block.
template <int K, int N>
__global__ void gcn_repack_B(const float* __restrict__ B,
                             float* __restrict__ Bp) {
    constexpr int KS = K / 4;
    int idx = blockIdx.x * blockDim.x + threadIdx.x;   // one float2 per thread
    if (idx >= (N / 16) * KS * 32) return;
    int lane = idx & 31;
    int ks   = (idx >> 5) % KS;
    int nt   = idx / (32 * KS);
    int hi   = lane >> 4;
    int n    = (nt << 4) + (lane & 15);
    int k    = (ks << 2) + (hi << 1);
    Bp[(size_t)idx * 2 + 0] = B[(size_t)k * N + n];
    Bp[(size_t)idx * 2 + 1] = B[(size_t)(k + 1) * N + n];
}

// ------------------------------------------------------------- WMMA fp32 GEMM
// D[M,N] = A[M,K] @ Bp (fragment-major), row-major A/D, fp32, exact.
// Block: 256 threads = 8 waves; one 16-row M-tile per block, one 16x16 output
// tile per wave. A-tile staged in LDS once per block (8x traffic reduction),
// B fragments stream as coalesced b64 loads out of L2.
// Fragment layouts per CDNA5 ISA 7.12.2 (fp32):
//   A 16x4 : lanes 0-15 hold K={k,k+1} of row m, lanes 16-31 K={k+2,k+3}
//   B 4x16 : lanes 0-15 hold rows K={k,k+1} at col n, lanes 16-31 K={k+2,k+3}
//   C/D    : reg r, lane l -> row (r + 8*(l>=16)), col (l&15)
template <int K, int N>
__global__ void gcn_gemm_wmma_f32(const float* __restrict__ A,
                                  const float* __restrict__ Bp,
                                  float* __restrict__ D) {
    constexpr int KS  = K / 4;     // WMMA k-steps
    constexpr int PAD = 4;         // LDS row pad (floats) to spread banks
    __shared__ float sA[16][K + PAD];

    const int lane = threadIdx.x & 31;
    const int wave = threadIdx.x >> 5;
    const int mt = blockIdx.x;              // 16-row tile
    const int nt = blockIdx.y * 8 + wave;   // 16-col tile (always < N/16)

    // ---- cooperative, coalesced A-tile load: 16 x K floats as float4 ------
    constexpr int QUADS = 16 * K / 4;       // float4 elements in the tile
    for (int q = threadIdx.x; q < QUADS; q += 256) {
        int row  = q / (K / 4);
        int col4 = q % (K / 4);
        float4 v = *((const float4*)(A + (size_t)((mt << 4) + row) * K) + col4);
        *(float4*)(&sA[row][col4 * 4]) = v;
    }
    __syncthreads();

    const int l15 = lane & 15;
    const int hi  = lane >> 4;              // half-wave select
    const int kh  = hi << 1;                // +0 or +2 inside each k-step

    // B fragment stream: 64 floats (256 B) per k-step, contiguous.
    const float* bPtr = Bp + ((size_t)nt * KS) * 64 + lane * 2;

    v8f acc = {};
#pragma unroll 8
    for (int ks = 0; ks < KS; ++ks) {
        v2f a = *(const v2f*)(&sA[l15][(ks << 2) + kh]);   // 8 B ds_load
        v2f b = *(const v2f*)(bPtr + (size_t)ks * 64);     // 8 B global b64
        acc = __builtin_amdgcn_wmma_f32_16x16x4_f32(
                  /*neg_a=*/false, a, /*neg_b=*/false, b,
                  /*c_mod=*/(short)0, acc, /*reuse_a=*/false, /*reuse_b=*/false);
    }

    const int n = (nt << 4) + l15;
    float* dRow = D + (size_t)((mt << 4) + (hi << 3)) * N + n;
#pragma unroll
    for (int r = 0; r < 8; ++r) {
        dRow[0] = acc[r];
        dRow += N;
    }
}

// ----------------------------------------------- self-loop init (+ opt. bias)
// agg[i][c] = h[i][c] * dinv[i]^2 (+ bias[c]); full overwrite, no atomics.
template <int C>
__global__ void gcn_self_init(const float* __restrict__ h,
                              const float* __restrict__ dinv,
                              const float* __restrict__ bias,   // may be null
                              float* __restrict__ agg) {
    long long idx = (long long)blockIdx.x * blockDim.x + threadIdx.x;
    if (idx >= (long long)N_NODES * C) return;
    int i = (int)(idx / C);                 // C is a power of two -> shifts
    int c = (int)(idx % C);
    float di = dinv[i];
    float v = h[idx] * di * di;
    if (bias) v += bias[c];
    agg[idx] = v;
}

// -------------------------------------------------- edge gather/scale/scatter
// One thread per (edge, 4-channel group): float4 gather from h[src] (h fits in
// the 192 MB L2), scale by norm = dinv[src]*dinv[dst], atomic scatter to dst.
template <int C>
__global__ void gcn_edge_scatter(const long long* __restrict__ src,
                                 const long long* __restrict__ dst,
                                 const float* __restrict__ dinv,
                                 const float* __restrict__ h,
                                 float* __restrict__ agg) {
    constexpr int G = C / 4;
    long long idx = (long long)blockIdx.x * blockDim.x + threadIdx.x;
    if (idx >= (long long)N_EDGES * G) return;
    int e = (int)(idx / G);                 // G power of two -> shifts
    int g = (int)(idx % G);
    int s = (int)src[e];
    int d = (int)dst[e];
    float nrm = dinv[s] * dinv[d];
    const float4 v = *((const float4*)(h + (size_t)s * C) + g);
    float* ap = agg + (size_t)d * C + (size_t)g * 4;
    atomicAdd(ap + 0, v.x * nrm);
    atomicAdd(ap + 1, v.y * nrm);
    atomicAdd(ap + 2, v.z * nrm);
    atomicAdd(ap + 3, v.w * nrm);
}

// ---------------------------------------------------------------- bias+ReLU
template <int C>
__global__ void gcn_bias_relu(float* __restrict__ a,
                              const float* __restrict__ bias) {
    long long idx = (long long)blockIdx.x * blockDim.x + threadIdx.x;
    if (idx >= (long long)N_NODES * C) return;
    int c = (int)(idx % C);
    a[idx] = fmaxf(a[idx] + bias[c], 0.0f);
}

// ---------------------------------------------------------------------------
extern "C" void kernel_launch(void* const* d_in, const int* in_sizes, int n_in,
                              void* d_out, int out_size, void* d_ws, size_t ws_size,
                              hipStream_t stream) {
    (void)in_sizes; (void)n_in; (void)out_size; (void)ws_size;

    const float*     x   = (const float*)d_in[0];
    const long long* ei  = (const long long*)d_in[1];   // int64 per reference
    const float*     W1  = (const float*)d_in[2];
    const float*     b1  = (const float*)d_in[3];
    const float*     W2  = (const float*)d_in[4];
    const float*     b2  = (const float*)d_in[5];
    float*           out = (float*)d_out;

    const long long* src = ei;              // edge_index[0]
    const long long* dst = ei + N_EDGES;    // edge_index[1]

    // Workspace layout (floats), 16B-aligned slices:
    float* ws   = (float*)d_ws;
    float* dinv = ws;                                        //  50,000 (pad 50,176)
    float* h1   = ws + 50176;                                //  50000*256
    float* agg1 = h1 + (size_t)N_NODES * HID_CH;             //  50000*256
    float* h2   = agg1 + (size_t)N_NODES * HID_CH;           //  50000*128
    float* Bp1  = h2 + (size_t)N_NODES * OUT_CH;             //  512*256
    float* Bp2  = Bp1 + (size_t)IN_CH * HID_CH;              //  256*128
    // total ~128.9 MB

    const dim3 blk(256);
    const int nodeBlocks = (N_NODES + 255) / 256;
    const int edgeBlocks = (N_EDGES + 255) / 256;

    // --- degree + symmetric normalization ---------------------------------
    gcn_fill<<<nodeBlocks, blk, 0, stream>>>(dinv, 1.0f, N_NODES);   // self-loop
    gcn_deg_accum<<<edgeBlocks, blk, 0, stream>>>(dst, dinv);
    gcn_rsqrt<<<nodeBlocks, blk, 0, stream>>>(dinv, N_NODES);

    // --- repack weights into WMMA-fragment-major order --------------------
    gcn_repack_B<IN_CH, HID_CH><<<(IN_CH * HID_CH / 2 + 255) / 256, blk, 0, stream>>>(W1, Bp1);
    gcn_repack_B<HID_CH, OUT_CH><<<(HID_CH * OUT_CH / 2 + 255) / 256, blk, 0, stream>>>(W2, Bp2);

    // --- layer 1: h1 = x @ W1 (fp32 WMMA, LDS-staged A) -------------------
    gcn_gemm_wmma_f32<IN_CH, HID_CH>
        <<<dim3(N_NODES / 16, HID_CH / 128), blk, 0, stream>>>(x, Bp1, h1);

    // self-loop messages (overwrite), then edge scatter, then bias+relu
    {
        long long n1 = (long long)N_NODES * HID_CH;
        int b = (int)((n1 + 255) / 256);
        gcn_self_init<HID_CH><<<b, blk, 0, stream>>>(h1, dinv, nullptr, agg1);

        long long ne = (long long)N_EDGES * (HID_CH / 4);
        int eb = (int)((ne + 255) / 256);
        gcn_edge_scatter<HID_CH><<<eb, blk, 0, stream>>>(src, dst, dinv, h1, agg1);

        gcn_bias_relu<HID_CH><<<b, blk, 0, stream>>>(agg1, b1);
    }

    // --- layer 2: h2 = h @ W2 (fp32 WMMA, LDS-staged A) -------------------
    gcn_gemm_wmma_f32<HID_CH, OUT_CH>
        <<<dim3(N_NODES / 16, OUT_CH / 128), blk, 0, stream>>>(agg1, Bp2, h2);

    {
        long long n2 = (long long)N_NODES * OUT_CH;
        int b = (int)((n2 + 255) / 256);
        gcn_self_init<OUT_CH><<<b, blk, 0, stream>>>(h2, dinv, b2, out);

        long long ne = (long long)N_EDGES * (OUT_CH / 4);
        int eb = (int)((ne + 255) / 256);
        gcn_edge_scatter<OUT_CH><<<eb, blk, 0, stream>>>(src, dst, dinv, h2, out);
    }
}